// KSetLayer_10797547782336
// MI455X (gfx1250) — compile-verified
//
#include <hip/hip_runtime.h>

#define DIM 64

typedef __attribute__((ext_vector_type(2))) float v2f;
typedef __attribute__((ext_vector_type(8))) float v8f;

// ---------------------------------------------------------------------------
// Kernel 1: fused dual GEMM.  self_out = x @ W1,  h2 = x @ W2
// One wave32 per 16-row tile of x; K=64 swept in 16 steps of V_WMMA_F32_16X16X4_F32.
//
// VGPR layout (CDNA5 ISA §7.12.2, 32-bit A 16x4):
//   A: lanes 0-15 hold row M=lane, {VGPR0,VGPR1} = K{0,1}; lanes 16-31 = K{2,3}.
//   B: VGPR0 = W[k0+2g][n], VGPR1 = W[k0+2g+1][n], n = ntile*16 + (lane&15).
//   C/D: VGPR v -> row (v + 8*(lane>=16)), col = lane&15 within the 16-col tile.
//
// W is staged in LDS pair-interleaved:  sWp[k/2][n] = {W[k][n], W[k+1][n]}
// so each B fragment is a single aligned 8-byte ds load (pairable into
// ds_load_2addr_b64 across N-tiles), with zero register shuffling.
// ---------------------------------------------------------------------------
__global__ __launch_bounds__(256) void gemm_dual(
    const float* __restrict__ x,
    const float* __restrict__ W1,
    const float* __restrict__ W2,
    float* __restrict__ self_out,
    float* __restrict__ h2,
    int n_nodes)
{
    __shared__ v2f sW1p[(DIM / 2) * DIM];   // 16 KB
    __shared__ v2f sW2p[(DIM / 2) * DIM];   // 16 KB

    for (int idx = threadIdx.x; idx < (DIM / 2) * DIM; idx += 256) {
        const int kk = idx >> 6;          // row-pair index
        const int n  = idx & (DIM - 1);   // column
        v2f p1, p2;
        p1.x = W1[(2 * kk    ) * DIM + n];
        p1.y = W1[(2 * kk + 1) * DIM + n];
        p2.x = W2[(2 * kk    ) * DIM + n];
        p2.y = W2[(2 * kk + 1) * DIM + n];
        sW1p[idx] = p1;
        sW2p[idx] = p2;
    }
    __syncthreads();

    const int wave = threadIdx.x >> 5;
    const int lane = threadIdx.x & 31;
    const int tile = blockIdx.x * 8 + wave;
    const int row0 = tile * 16;
    if (row0 >= n_nodes) return;

    const int m = lane & 15;     // row within tile for A; col within N-tile for B/D
    const int g = lane >> 4;     // lane-group: selects K pair {0,1} vs {2,3}

    if (row0 + 16 <= n_nodes) {
        // ---- full-tile WMMA path (wave-uniform branch, EXEC all ones) ----
        v8f acc1[4], acc2[4];
        #pragma unroll
        for (int nt = 0; nt < 4; ++nt) {
            acc1[nt] = (v8f){0.f,0.f,0.f,0.f,0.f,0.f,0.f,0.f};
            acc2[nt] = (v8f){0.f,0.f,0.f,0.f,0.f,0.f,0.f,0.f};
        }

        const float* xrow = x + (size_t)(row0 + m) * DIM;

        #pragma unroll
        for (int k0 = 0; k0 < DIM; k0 += 4) {
            const int kp = (k0 >> 1) + g;               // LDS row-pair index
            v2f a = *(const v2f*)(xrow + k0 + 2 * g);   // 8-byte coalesced load
            const v2f* b1p = sW1p + kp * DIM + m;
            const v2f* b2p = sW2p + kp * DIM + m;
            #pragma unroll
            for (int nt = 0; nt < 4; ++nt) {
                const v2f b1 = b1p[nt * 16];            // single ds_load_b64
                const v2f b2 = b2p[nt * 16];
                acc1[nt] = __builtin_amdgcn_wmma_f32_16x16x4_f32(
                    false, a, false, b1, (short)0, acc1[nt], false, false);
                acc2[nt] = __builtin_amdgcn_wmma_f32_16x16x4_f32(
                    false, a, false, b2, (short)0, acc2[nt], false, false);
            }
        }

        #pragma unroll
        for (int nt = 0; nt < 4; ++nt) {
            #pragma unroll
            for (int v = 0; v < 8; ++v) {
                const int row = row0 + v + 8 * g;
                const int col = nt * 16 + m;
                self_out[(size_t)row * DIM + col] = acc1[nt][v];
                h2      [(size_t)row * DIM + col] = acc2[nt][v];
            }
        }
    } else {
        // ---- ragged tail tile (not hit for N=100000): scalar fallback ----
        const int row = row0 + lane;
        if (row < n_nodes) {
            const float* xr = x + (size_t)row * DIM;
            for (int n = 0; n < DIM; ++n) {
                float s1 = 0.f, s2 = 0.f;
                for (int k = 0; k < DIM; ++k) {
                    const float xv = xr[k];
                    s1 += xv * sW1p[(k >> 1) * DIM + n][k & 1];
                    s2 += xv * sW2p[(k >> 1) * DIM + n][k & 1];
                }
                self_out[(size_t)row * DIM + n] = s1;
                h2      [(size_t)row * DIM + n] = s2;
            }
        }
    }
}

// ---------------------------------------------------------------------------
// Kernel 2: edge scatter.  out[dst] += h2[src]  (16 threads/edge, float4 each).
// Gather is a contiguous 256B row read; atomics are 4 contiguous f32 adds,
// all L2-resident (working set < 192MB L2).
// ---------------------------------------------------------------------------
__global__ __launch_bounds__(256) void scatter_add(
    const int* __restrict__ ei,      // [2, E] row-major
    const float* __restrict__ h2,
    float* __restrict__ out,
    int n_edges)
{
    const long long t = (long long)blockIdx.x * 256 + threadIdx.x;
    const int e = (int)(t >> 4);
    if (e >= n_edges) return;
    const int c   = (int)(t & 15);
    const int src = ei[e];
    const int dst = ei[n_edges + e];

    const float4 v = *(const float4*)(h2 + (size_t)src * DIM + c * 4);
    float* o = out + (size_t)dst * DIM + c * 4;
    atomicAdd(o + 0, v.x);
    atomicAdd(o + 1, v.y);
    atomicAdd(o + 2, v.z);
    atomicAdd(o + 3, v.w);
}

// ---------------------------------------------------------------------------
// Kernel 3: in-place ReLU (float4 per thread).
// ---------------------------------------------------------------------------
__global__ __launch_bounds__(256) void relu_k(float* __restrict__ out, int n4)
{
    const int t = blockIdx.x * 256 + threadIdx.x;
    if (t >= n4) return;
    float4* p = (float4*)out + t;
    float4 v = *p;
    v.x = fmaxf(v.x, 0.f);
    v.y = fmaxf(v.y, 0.f);
    v.z = fmaxf(v.z, 0.f);
    v.w = fmaxf(v.w, 0.f);
    *p = v;
}

// ---------------------------------------------------------------------------
extern "C" void kernel_launch(void* const* d_in, const int* in_sizes, int n_in,
                              void* d_out, int out_size, void* d_ws, size_t ws_size,
                              hipStream_t stream)
{
    const float* x  = (const float*)d_in[0];
    const int*   ei = (const int*)  d_in[1];
    const float* W1 = (const float*)d_in[2];
    const float* W2 = (const float*)d_in[3];
    float* out = (float*)d_out;
    float* h2  = (float*)d_ws;           // N * 64 floats of scratch

    const int n_nodes = in_sizes[0] / DIM;
    const int n_edges = in_sizes[1] / 2;

    // Phase 1: self_out = x@W1 -> d_out (overwrite), h2 = x@W2 -> scratch
    const int tiles   = (n_nodes + 15) / 16;
    const int blocksA = (tiles + 7) / 8;                 // 8 waves per block
    gemm_dual<<<blocksA, 256, 0, stream>>>(x, W1, W2, out, h2, n_nodes);

    // Phase 2: out[dst] += h2[src] over all edges
    const long long tB = (long long)n_edges * 16;
    const int blocksB = (int)((tB + 255) / 256);
    scatter_add<<<blocksB, 256, 0, stream>>>(ei, h2, out, n_edges);

    // Phase 3: ReLU in place
    const int n4 = n_nodes * DIM / 4;
    relu_k<<<(n4 + 255) / 256, 256, 0, stream>>>(out, n4);
}